// Model_42434276884878
// MI455X (gfx1250) — compile-verified
//
#include <hip/hip_runtime.h>
#include <hip/hip_bf16.h>
#include <math.h>

// Problem constants (from reference)
#define BB   64
#define LL   8192
#define CC   8
#define MA_K 25          // moving-average kernel
#define FREQ 96
#define HALF 49          // FREQ/2 + 1
#define WW   8097        // LL - FREQ + 1 : DFT length and out time dim
#define OC   101         // 49 real + 49 imag + seas + trend + raw

typedef float v2f __attribute__((ext_vector_type(2)));
typedef float v8f __attribute__((ext_vector_type(8)));

// ---------------------------------------------------------------------------
// Workspace layout (floats):
//   [0              .. 16384)              twiddle table float2[WW] (padded)
//   [16384          .. 16384+2*WW*64)      X = base DFT  float2[WW*64] (k-major)
//   [16384+2*WW*64  .. +LL*64)             sT = seasonal transposed [t][b]
// total ~= 6.31 MB
// ---------------------------------------------------------------------------
#define WS_TW 0
#define WS_X  16384
#define WS_ST (16384 + 2 * WW * 64)

// ---- Kernel 1: twiddle table tw[j] = (cos, sin)(2*pi*j/W) ------------------
__global__ __launch_bounds__(256) void k_twiddle(float* __restrict__ ws) {
  int j = blockIdx.x * 256 + threadIdx.x;
  if (j < WW) {
    float a = (float)(6.283185307179586 * (double)j / (double)WW);
    float s, c;
    sincosf(a, &s, &c);
    ((float2*)(ws + WS_TW))[j] = make_float2(c, s);
  }
}

// ---- Kernel 2: series decomposition --------------------------------------
// trend = 25-tap edge-replicated moving average of s = x[:,:,7]
// writes seasonal transposed sT[t][b] and output channels 98..100 directly.
__global__ __launch_bounds__(256) void k_decomp(const float* __restrict__ x,
                                                float* __restrict__ ws,
                                                float* __restrict__ out) {
  __shared__ float sh[256 + MA_K - 1];  // 280 floats, 12-elem halo each side
  const int tid = threadIdx.x;
  const int b   = blockIdx.y;
  const int t0  = blockIdx.x * 256;
  float* sT = ws + WS_ST;

  for (int i = tid; i < 256 + MA_K - 1; i += 256) {
    int t = t0 - 12 + i;                       // edge replication = clamp
    t = t < 0 ? 0 : (t > LL - 1 ? LL - 1 : t);
    sh[i] = x[((size_t)b * LL + t) * CC + (CC - 1)];
  }
  __syncthreads();

  const int t = t0 + tid;
  float sum = 0.0f;
#pragma unroll
  for (int j = 0; j < MA_K; ++j) sum += sh[tid + j];
  const float trend = sum / (float)MA_K;
  const float sv    = sh[tid + 12];
  const float seas  = sv - trend;

  sT[(size_t)t * 64 + b] = seas;               // transposed for DFT B-operand
  if (t >= FREQ - 1) {
    size_t o = ((size_t)b * WW + (t - (FREQ - 1))) * OC;
    out[o + 98]  = seas;
    out[o + 99]  = trend;
    out[o + 100] = sv;
  }
}

// ---- Kernel 3: base DFT via V_WMMA_F32_16X16X4_F32 -----------------------
// X[k][b] = sum_w s[b][w] * e^{-2*pi*i*k*w/W}, k in [0,WW), all 64 batches.
// A-tile (16k x 4w) built per-lane from the LDS twiddle table with an
// incremental (k*w mod W) index (no transcendentals in the hot loop).
// Block = 8 waves: 2 k-subtiles x 4 b-subtiles  => 32 k x 64 b per block.
__global__ __launch_bounds__(256) void k_basedft(const float* __restrict__ ws) {
  __shared__ float2 tw[WW];                    // 64,776 B of 320 KB LDS
  const int tid = threadIdx.x;
  const float2* twg = (const float2*)(ws + WS_TW);
  for (int i = tid; i < WW; i += 256) tw[i] = twg[i];
  __syncthreads();

  const float* sT = ws + WS_ST;
  float2*      X  = (float2*)(ws + WS_X);

  const int lane = tid & 31;
  const int wv   = tid >> 5;
  const int m    = lane & 15;                  // M (A) / N (B,D) lane index
  const int h    = lane >> 4;                  // half-wave selector
  const int ktile = blockIdx.x * 32 + (wv & 1) * 16;
  const int btile = (wv >> 1) * 16;
  const int bcol  = btile + m;

  int kk = ktile + m;
  if (kk > WW - 1) kk = WW - 1;                // clamp (guarded at store)
  // A layout (ISA 7.12.2): VGPR j, lane half h -> K = 2h + j, M = m
  int idx0 = (kk * (2 * h + 0)) % WW;
  int idx1 = (kk * (2 * h + 1)) % WW;
  const int step = (4 * kk) % WW;              // w advances by 4 per iter

  v8f accC = {};                               // sum s*cos -> Re
  v8f accS = {};                               // sum s*sin -> -Im

  for (int wb = 0; wb < WW; wb += 4) {
    float2 t0 = tw[idx0];                      // ds_load_b64: (cos,sin)
    float2 t1 = tw[idx1];
    idx0 += step; if (idx0 >= WW) idx0 -= WW;
    idx1 += step; if (idx1 >= WW) idx1 -= WW;

    // B layout (symmetric to A): VGPR j, half h -> K = 2h + j, N = m
    const int w0 = wb + 2 * h;
    float b0 = (w0     < WW) ? sT[(size_t)w0       * 64 + bcol] : 0.0f;
    float b1 = (w0 + 1 < WW) ? sT[(size_t)(w0 + 1) * 64 + bcol] : 0.0f;

    v2f Ac = { t0.x, t1.x };
    v2f As = { t0.y, t1.y };
    v2f Bv = { b0, b1 };
    accC = __builtin_amdgcn_wmma_f32_16x16x4_f32(false, Ac, false, Bv,
                                                 (short)0, accC, false, false);
    accS = __builtin_amdgcn_wmma_f32_16x16x4_f32(false, As, false, Bv,
                                                 (short)0, accS, false, false);
  }

  // D layout: lane (m,h), elem v -> k = ktile + 8h + v, b = bcol
#pragma unroll
  for (int v = 0; v < 8; ++v) {
    const int ko = ktile + 8 * h + v;
    if (ko < WW) {
      float2 r;
      r.x = accC[v];
      r.y = -accS[v];                          // e^{-i t} = cos - i sin
      X[(size_t)ko * 64 + bcol] = r;           // 128B-contiguous per half-wave
    }
  }
}

// ---- Kernel 4: sliding DFT across shifts f = 0..48 -----------------------
// X_{f+1}[k] = e^{+2*pi*i*k/W} * (X_f[k] + (s[f+W] - s[f]))
// Results staged in LDS, then written as contiguous 98-float rows.
__global__ __launch_bounds__(64) void k_slide(const float* __restrict__ ws,
                                              float* __restrict__ out) {
  __shared__ float tile[64 * 98];              // 24.5 KB
  __shared__ float delta[HALF - 1];
  const int tid   = threadIdx.x;
  const int b     = blockIdx.y;
  const int kbase = blockIdx.x * 64;

  const float*  sT  = ws + WS_ST;
  const float2* twg = (const float2*)(ws + WS_TW);
  const float2* X   = (const float2*)(ws + WS_X);

  if (tid < HALF - 1)
    delta[tid] = sT[(size_t)(tid + WW) * 64 + b] - sT[(size_t)tid * 64 + b];
  __syncthreads();

  const int k = kbase + tid;
  if (k < WW) {
    const float2 t  = twg[k];
    const float  cr = t.x, ci = t.y;           // e^{+2*pi*i*k/W}
    const float2 x0 = X[(size_t)k * 64 + b];
    float xr = x0.x, xi = x0.y;
    float* row = &tile[tid * 98];
    for (int f = 0; f < HALF; ++f) {
      row[f]        = xr;                      // spec.real channel f
      row[HALF + f] = xi;                      // spec.imag channel 49+f
      if (f < HALF - 1) {
        const float ar = xr + delta[f];
        const float nr = ar * cr - xi * ci;
        const float ni = ar * ci + xi * cr;
        xr = nr; xi = ni;
      }
    }
  }
  __syncthreads();

  // coalesced writeback: 98 consecutive floats per output row (stride 101)
  for (int r = 0; r < 64; ++r) {
    const int ko = kbase + r;
    if (ko >= WW) break;
    const size_t base = ((size_t)b * WW + ko) * OC;
    for (int c = tid; c < 98; c += 64)
      out[base + c] = tile[r * 98 + c];
  }
}

// ---------------------------------------------------------------------------
extern "C" void kernel_launch(void* const* d_in, const int* in_sizes, int n_in,
                              void* d_out, int out_size, void* d_ws, size_t ws_size,
                              hipStream_t stream) {
  (void)in_sizes; (void)n_in; (void)out_size; (void)ws_size;
  const float* x  = (const float*)d_in[0];
  float* out      = (float*)d_out;
  float* ws       = (float*)d_ws;              // needs ~6.31 MB

  k_twiddle<<<dim3((WW + 255) / 256), 256, 0, stream>>>(ws);
  k_decomp <<<dim3(LL / 256, BB),     256, 0, stream>>>(x, ws, out);
  k_basedft<<<dim3((WW + 31) / 32),   256, 0, stream>>>(ws);
  k_slide  <<<dim3((WW + 63) / 64, BB), 64, 0, stream>>>(ws, out);
}